// SupConLoss_84318797955809
// MI455X (gfx1250) — compile-verified
//
#include <hip/hip_runtime.h>
#include <hip/hip_bf16.h>
#include <stdint.h>

typedef __attribute__((ext_vector_type(2))) float v2f;
typedef __attribute__((ext_vector_type(4))) float v4f;
typedef __attribute__((ext_vector_type(8))) float v8f;

typedef unsigned int u32x4 __attribute__((ext_vector_type(4)));
typedef int          i32x4 __attribute__((ext_vector_type(4)));
typedef int          i32x8 __attribute__((ext_vector_type(8)));

typedef __attribute__((address_space(3))) void lds_void_t;

#define SC_TEMP      0.07f
#define SC_EPS       1e-8f
#define LDS_STRIDE   260      // 256 + 4 pad -> conflict-free b64 LDS reads
#define KDIM         256      // feature dim (reference: 256)
#define KSTEPS       64       // KDIM / 4 (WMMA f32 16x16x4)
#define TILE_FLOATS  (16 * LDS_STRIDE)

#if defined(__has_builtin)
#if __has_builtin(__builtin_amdgcn_tensor_load_to_lds) && \
    __has_builtin(__builtin_amdgcn_s_wait_tensorcnt)
#define USE_TDM 1
#endif
#endif
#ifndef USE_TDM
#define USE_TDM 0
#endif

#if USE_TDM
// Issue a TDM DMA: 16-row x 256-float tile of F (rows col_base..col_base+15)
// into LDS at lds_byte_addr, with 4-dword padding every 256 dwords so the
// resulting LDS row stride is LDS_STRIDE floats (bank-conflict-free reads).
__device__ __forceinline__ void tdm_issue_tile(const float* __restrict__ F,
                                               int col_base, int N,
                                               unsigned lds_byte_addr) {
  unsigned long long ga =
      (unsigned long long)(uintptr_t)(F + (size_t)col_base * KDIM);

  u32x4 g0;
  g0.x = 1u;                                   // count=1 (valid user D#)
  g0.y = lds_byte_addr;                        // lds_addr
  g0.z = (unsigned)(ga & 0xffffffffu);         // global_addr[31:0]
  g0.w = (unsigned)((ga >> 32) & 0x01ffffffu)  // global_addr[56:32]
         | (2u << 30);                         // type=2 ("image")

  i32x8 g1;
  g1[0] = (int)((2u << 16)        // data_size = 4 bytes
                | (1u << 20)      // pad_enable
                | (7u << 22)      // pad_interval: 256 dwords
                | (3u << 25));    // pad_amount: 4 dwords -> stride 260 floats
  g1[1] = (int)(((unsigned)KDIM & 0xffffu) << 16);        // tensor_dim0 lo16
  g1[2] = (int)(((unsigned)N & 0xffffu) << 16);           // dim0 hi | dim1 lo
  g1[3] = (int)(((unsigned)KDIM & 0xffffu) << 16);        // dim1 hi | tile_dim0
  g1[4] = 16;                                             // tile_dim1=16 rows
  g1[5] = KDIM;                                           // dim0_stride lo32
  g1[6] = 0;                                              // stride hi | d1_stride
  g1[7] = 0;

  i32x4 gz = {0, 0, 0, 0};
#if __clang_major__ >= 23
  i32x8 gz8 = {0, 0, 0, 0, 0, 0, 0, 0};
  __builtin_amdgcn_tensor_load_to_lds(g0, g1, gz, gz, gz8, 0);
#else
  __builtin_amdgcn_tensor_load_to_lds(g0, g1, gz, gz, 0);
#endif
}
#endif  // USE_TDM

// ---------------------------------------------------------------------------
// Main fused kernel: one wave per 16-row block; TDM double-buffered column
// tiles; online softmax; never materializes the NxN similarity matrix.
// ---------------------------------------------------------------------------
__global__ __launch_bounds__(256) void supcon_main_kernel(
    const float* __restrict__ F, const int* __restrict__ labels,
    int N, int B, int ntiles, float* __restrict__ partial) {
  __shared__ float bt[2][TILE_FLOATS];

  const int lane        = threadIdx.x & 31;
  const int wave        = threadIdx.x >> 5;
  const int wave_global = blockIdx.x * 8 + wave;
  const int row_base    = wave_global * 16;

  const int lm     = lane & 15;        // col-within-tile / row-within-Afrag
  const int khalf  = lane >> 4;        // 0: K=0,1  1: K=2,3 (A & B frag layout)
  const int koff   = khalf * 2;
  const int rowoff = khalf * 8;        // C/D layout: upper half-lanes hold M+8

  // --- A fragments: 16 rows x 256 K, 64 x float2 per lane (registers) ------
  v2f afrag[KSTEPS];
  {
    const float* arow = F + (size_t)(row_base + lm) * KDIM;
#pragma unroll
    for (int kk = 0; kk < KSTEPS; ++kk)
      afrag[kk] = *(const v2f*)(arow + 4 * kk + koff);
  }

  // --- labels for the 8 rows this lane accumulates -------------------------
  int labr[8];
#pragma unroll
  for (int j = 0; j < 8; ++j)
    labr[j] = labels[(row_base + rowoff + j) % B];

  // --- online softmax running state per owned row --------------------------
  float m[8], s[8], p[8], c[8];
#pragma unroll
  for (int j = 0; j < 8; ++j) { m[j] = -3.0e38f; s[j] = 0.f; p[j] = 0.f; c[j] = 0.f; }

  const float invT = 1.0f / SC_TEMP;

#if USE_TDM
  const unsigned lds_b0 =
      (unsigned)(size_t)(lds_void_t*)&bt[0][0];
  const unsigned lds_b1 =
      (unsigned)(size_t)(lds_void_t*)&bt[1][0];
  // Pre-load tile 0 into buffer 0 (wave 0 drives all DMA for the block).
  if (wave == 0) tdm_issue_tile(F, 0, N, lds_b0);
#else
  const int tcol = threadIdx.x >> 4;        // 0..15: col staged by this thread
  const int tk   = (threadIdx.x & 15) * 16; // 16-float K chunk
  {
    const float* src = F + (size_t)tcol * KDIM + tk;
#pragma unroll
    for (int j = 0; j < 4; ++j)
      *(v4f*)&bt[0][tcol * LDS_STRIDE + tk + 4 * j] = *(const v4f*)(src + 4 * j);
  }
#endif

  for (int ct = 0; ct < ntiles; ++ct) {
    const int b = ct & 1;
    const int col_base = ct * 16;

#if USE_TDM
    // Publish buf[b]: DMA issued last iteration (or preloop) must be done.
    if (wave == 0) __builtin_amdgcn_s_wait_tensorcnt((unsigned short)0);
    __syncthreads();  // all waves see buf[b]; all reads of buf[1-b] retired
    // Kick DMA for the next tile into the other buffer; overlaps the WMMAs.
    if (wave == 0 && ct + 1 < ntiles)
      tdm_issue_tile(F, col_base + 16, N, b ? lds_b0 : lds_b1);
#else
    __syncthreads();
#endif

    const float* __restrict__ btb = &bt[b][0];

    // --- 16x16 tile of F * F^T via V_WMMA_F32_16X16X4_F32 ------------------
    v8f a0 = {}, a1 = {}, a2 = {}, a3 = {};
#pragma unroll
    for (int kk = 0; kk < KSTEPS; kk += 4) {
      v2f b0 = *(const v2f*)&btb[lm * LDS_STRIDE + 4 * (kk + 0) + koff];
      v2f b1 = *(const v2f*)&btb[lm * LDS_STRIDE + 4 * (kk + 1) + koff];
      v2f b2 = *(const v2f*)&btb[lm * LDS_STRIDE + 4 * (kk + 2) + koff];
      v2f b3 = *(const v2f*)&btb[lm * LDS_STRIDE + 4 * (kk + 3) + koff];
      a0 = __builtin_amdgcn_wmma_f32_16x16x4_f32(false, afrag[kk + 0], false, b0,
                                                 (short)0, a0, false, false);
      a1 = __builtin_amdgcn_wmma_f32_16x16x4_f32(false, afrag[kk + 1], false, b1,
                                                 (short)0, a1, false, false);
      a2 = __builtin_amdgcn_wmma_f32_16x16x4_f32(false, afrag[kk + 2], false, b2,
                                                 (short)0, a2, false, false);
      a3 = __builtin_amdgcn_wmma_f32_16x16x4_f32(false, afrag[kk + 3], false, b3,
                                                 (short)0, a3, false, false);
    }

    v8f acc;
#pragma unroll
    for (int j = 0; j < 8; ++j) acc[j] = a0[j] + a1[j] + a2[j] + a3[j];

    // --- online softmax + positive-mask accumulation -----------------------
    const int gcol = col_base + lm;
    const int labc = labels[gcol % B];
#pragma unroll
    for (int j = 0; j < 8; ++j) {
      const int grow = row_base + rowoff + j;
      const float sim = acc[j] * invT;
      const bool self = (gcol == grow);
      const float nm = fmaxf(m[j], sim);           // max includes self (ref)
      const float e  = self ? 0.0f : __expf(sim - nm);
      s[j] = s[j] * __expf(m[j] - nm) + e;         // denom excludes self
      m[j] = nm;
      if ((labr[j] == labc) && !self) { p[j] += sim; c[j] += 1.0f; }
    }

#if !USE_TDM
    __syncthreads();
    if (ct + 1 < ntiles) {
      const float* src = F + (size_t)(col_base + 16 + tcol) * KDIM + tk;
#pragma unroll
      for (int j = 0; j < 4; ++j)
        *(v4f*)&bt[(ct + 1) & 1][tcol * LDS_STRIDE + tk + 4 * j] =
            *(const v4f*)(src + 4 * j);
    }
#endif
  }

  // --- reduce across the 16 lanes of each row group (xor 1,2,4,8) ----------
#pragma unroll
  for (int j = 0; j < 8; ++j) {
#pragma unroll
    for (int off = 1; off <= 8; off <<= 1) {
      const float om = __shfl_xor(m[j], off, 32);
      const float os = __shfl_xor(s[j], off, 32);
      const float op = __shfl_xor(p[j], off, 32);
      const float oc = __shfl_xor(c[j], off, 32);
      const float nm = fmaxf(m[j], om);
      s[j] = s[j] * __expf(m[j] - nm) + os * __expf(om - nm);
      m[j] = nm;
      p[j] += op;
      c[j] += oc;
    }
  }

  // per-lane sum of mean_log_prob_pos for its 8 rows (identical within halves)
  float t = 0.0f;
#pragma unroll
  for (int j = 0; j < 8; ++j) {
    const float denom = fmaxf(s[j], SC_EPS);
    const float mlpp =
        (c[j] < 0.5f) ? 0.0f : (p[j] / c[j] - (m[j] + __logf(denom)));
    t += mlpp;
  }
  const float other   = __shfl_xor(t, 16, 32);  // other 8 rows of the block
  const float wavesum = t + other;
  if (lane == 0) partial[wave_global] = wavesum;
}

// ---------------------------------------------------------------------------
// Deterministic final reduction: loss = -(T/BASE_T) * mean(mlpp) = -sum/N
// ---------------------------------------------------------------------------
__global__ __launch_bounds__(256) void supcon_reduce_kernel(
    const float* __restrict__ partial, float* __restrict__ out, int nparts,
    float scale) {
  __shared__ float buf[256];
  const int t = threadIdx.x;
  float v = 0.0f;
  for (int i = t; i < nparts; i += 256) v += partial[i];
  buf[t] = v;
  __syncthreads();
  for (int st = 128; st > 0; st >>= 1) {
    if (t < st) buf[t] += buf[t + st];
    __syncthreads();
  }
  if (t == 0) out[0] = buf[0] * scale;
}

extern "C" void kernel_launch(void* const* d_in, const int* in_sizes, int n_in,
                              void* d_out, int out_size, void* d_ws,
                              size_t ws_size, hipStream_t stream) {
  const float* F      = (const float*)d_in[0];
  const int*   labels = (const int*)d_in[1];

  const int B = in_sizes[1];  // 2048 labels
  const int N = 2 * B;        // N_VIEWS = 2 -> 4096 rows (KDIM fixed at 256)

  const int nwaves  = N / 16;            // one wave per 16-row block (256)
  const int nblocks = (nwaves + 7) / 8;  // 8 waves per 256-thread block
  const int ntiles  = N / 16;            // column tiles

  float* partial = (float*)d_ws;

  supcon_main_kernel<<<nblocks, 256, 0, stream>>>(F, labels, N, B, ntiles,
                                                  partial);
  supcon_reduce_kernel<<<1, 256, 0, stream>>>(partial, (float*)d_out, nwaves,
                                              -1.0f / (float)N);
}